// LSTMTagger_7413113553595
// MI455X (gfx1250) — compile-verified
//
#include <hip/hip_runtime.h>
#include <math.h>

#define T_LEN 2048
#define M_MORP 6144
#define EMB 100
#define DPDIM 64
#define FEATN 36
#define HID 128
#define IN0 300          // 2*EMB + DPDIM + FEATN
#define IN1 256          // 2*HID
#define G4 512           // 4*HID gate rows
#define TAGS 50

typedef float v2f __attribute__((ext_vector_type(2)));
typedef float v8f __attribute__((ext_vector_type(8)));

// ------------------------------------------------------------------
// 1) Deterministic segment-sum: segment_ids are sorted, so each block
//    (one per token t) binary-searches its morpheme range and sums.
// ------------------------------------------------------------------
__global__ void segsum_kernel(const float* __restrict__ Wwv,
                              const int* __restrict__ morp,
                              const int* __restrict__ seg,
                              float* __restrict__ tok) {
  __shared__ int sLo, sHi;
  const int t = blockIdx.x;
  if (threadIdx.x == 0) {
    int lo = 0, hi = M_MORP;
    while (lo < hi) { int mid = (lo + hi) >> 1; if (seg[mid] < t) lo = mid + 1; else hi = mid; }
    sLo = lo;
    lo = 0; hi = M_MORP;
    while (lo < hi) { int mid = (lo + hi) >> 1; if (seg[mid] < t + 1) lo = mid + 1; else hi = mid; }
    sHi = lo;
  }
  __syncthreads();
  const int e = threadIdx.x;
  if (e < EMB) {
    float s = 0.0f;
    for (int m = sLo; m < sHi; ++m) s += Wwv[(long)morp[m] * EMB + e];
    tok[(long)t * EMB + e] = s;
  }
}

// ------------------------------------------------------------------
// 2) Assemble the 300-wide LSTM input row per token:
//    [tok_sums | pred_vec bcast | dp_emb | feat]
// ------------------------------------------------------------------
__global__ void build_input_kernel(const float* __restrict__ tok,
                                   const int* __restrict__ pred_idx,
                                   const float* __restrict__ dp_table,
                                   const int* __restrict__ dp_in,
                                   const float* __restrict__ feat,
                                   float* __restrict__ inp) {
  const int t = blockIdx.x;
  const int c = threadIdx.x;          // block of 320, cols 0..299 valid
  if (c >= IN0) return;
  const int pred = *pred_idx;
  float v;
  if (c < EMB)                       v = tok[(long)t * EMB + c];
  else if (c < 2 * EMB)              v = tok[(long)pred * EMB + (c - EMB)];
  else if (c < 2 * EMB + DPDIM)      v = dp_table[(long)dp_in[t] * DPDIM + (c - 2 * EMB)];
  else                               v = feat[(long)t * FEATN + (c - 2 * EMB - DPDIM)];
  inp[(long)t * IN0 + c] = v;
}

// ------------------------------------------------------------------
// 3) fp32 WMMA GEMM:  C[M x nValid] = A[M x K] * Bt[nValid x K]^T + bias
//    One wave per 16x16 tile, V_WMMA_F32_16X16X4_F32 over K.
//    A frag : lane m = lane&15, holds K = {kq, kq+1}, kq = (lane>>4)*2
//    B frag : lane n = lane&15, same K packing (B[k][n] = Bt[n][k])
//    D frag : acc[r] = D[r + 8*(lane>>4)][lane&15]
//    N-tail handling is branchless: out-of-range lanes read B row 0
//    (memory-safe) — their accumulator columns are simply never stored,
//    since lane n only ever contributes to output column n.
// ------------------------------------------------------------------
__global__ void gemm_wmma_f32(const float* __restrict__ A,
                              const float* __restrict__ Bt,
                              const float* __restrict__ bias,
                              float* __restrict__ C,
                              int K, int ldc, int nValid) {
  const int tileM = blockIdx.x * 16;
  const int tileN = blockIdx.y * 16;
  const int lane  = threadIdx.x;          // 0..31, wave32
  const int half  = lane >> 4;            // 0 or 1 -> K sub-offset
  const int mr    = lane & 15;
  const int n     = tileN + (lane & 15);
  const bool nOk  = (n < nValid);
  const int nClamp = nOk ? n : 0;         // branchless, memory-safe

  const float2* __restrict__ arow = (const float2*)(A + (long)(tileM + mr) * K);
  const float2* __restrict__ brow = (const float2*)(Bt + (long)nClamp * K);

  v8f acc = {0.f, 0.f, 0.f, 0.f, 0.f, 0.f, 0.f, 0.f};
  const int K2 = K >> 1;                  // K in float2 units (K % 4 == 0)
  for (int k2 = half; k2 < K2; k2 += 2) { // k2 = kb/2 + half
    const float2 av = arow[k2];
    const float2 bv = brow[k2];
    v2f a; a.x = av.x; a.y = av.y;
    v2f b; b.x = bv.x; b.y = bv.y;
    acc = __builtin_amdgcn_wmma_f32_16x16x4_f32(false, a, false, b,
                                                (short)0, acc, false, false);
  }

  if (nOk) {
    const float bv = bias[n];
    const int mhalf = half * 8;
#pragma unroll
    for (int r = 0; r < 8; ++r) {
      const int m = tileM + r + mhalf;
      C[(long)m * ldc + n] = acc[r] + bv;
    }
  }
}

// ------------------------------------------------------------------
// 4) Sequential LSTM scan. grid=2 (block 0 = forward, block 1 = backward),
//    512 threads: thread j owns gate-row j; W_hh row cached in 128 VGPRs;
//    h lives in LDS, read as float4 broadcasts (1 ds_load_b128 per 4 FMAs).
//    xp for the next step is prefetched to hide global latency.
// ------------------------------------------------------------------
__global__ void __launch_bounds__(512, 1)
lstm_scan2(const float* __restrict__ xpF, const float* __restrict__ whhF,
           const float* __restrict__ xpB, const float* __restrict__ whhB,
           float* __restrict__ hout, int Tlen) {
  __shared__ float4 lds_h4[HID / 4];
  __shared__ float  lds_g[G4];
  float* lds_h = (float*)lds_h4;

  const int dir = blockIdx.x;               // 0 = fwd, 1 = bwd
  const float* __restrict__ xp  = dir ? xpB  : xpF;
  const float* __restrict__ whh = dir ? whhB : whhF;
  const int colOff = dir ? HID : 0;

  const int j = threadIdx.x;                // 0..511 gate row

  // cache W_hh row j in registers (128 floats)
  float wrow[HID];
  const float4* __restrict__ w4 = (const float4*)(whh + (long)j * HID);
#pragma unroll
  for (int q = 0; q < HID / 4; ++q) {
    float4 v = w4[q];
    wrow[4 * q + 0] = v.x; wrow[4 * q + 1] = v.y;
    wrow[4 * q + 2] = v.z; wrow[4 * q + 3] = v.w;
  }

  if (j < HID) lds_h[j] = 0.0f;
  float c = 0.0f;
  __syncthreads();

  const int dt = dir ? -1 : 1;
  int tt = dir ? (Tlen - 1) : 0;
  float xnext = xp[(long)tt * G4 + j];

  for (int s = 0; s < Tlen; ++s) {
    const float xv = xnext;
    const int tn = tt + dt;
    if (s + 1 < Tlen) xnext = xp[(long)tn * G4 + j];   // prefetch

    float g = xv;
#pragma unroll
    for (int q = 0; q < HID / 4; ++q) {
      float4 hv = lds_h4[q];
      g = fmaf(hv.x, wrow[4 * q + 0], g);
      g = fmaf(hv.y, wrow[4 * q + 1], g);
      g = fmaf(hv.z, wrow[4 * q + 2], g);
      g = fmaf(hv.w, wrow[4 * q + 3], g);
    }
    lds_g[j] = g;
    __syncthreads();

    if (j < HID) {
      const float gi = lds_g[j];
      const float gf = lds_g[HID + j];
      const float gc = lds_g[2 * HID + j];
      const float go = lds_g[3 * HID + j];
      const float ig = 1.0f / (1.0f + expf(-gi));
      const float fg = 1.0f / (1.0f + expf(-gf));
      const float og = 1.0f / (1.0f + expf(-go));
      c = fg * c + ig * tanhf(gc);
      const float h = og * tanhf(c);
      lds_h[j] = h;
      hout[(long)tt * IN1 + colOff + j] = h;
    }
    __syncthreads();
    tt = tn;
  }
}

// ------------------------------------------------------------------
extern "C" void kernel_launch(void* const* d_in, const int* in_sizes, int n_in,
                              void* d_out, int out_size, void* d_ws, size_t ws_size,
                              hipStream_t stream) {
  (void)in_sizes; (void)n_in; (void)out_size; (void)ws_size;

  const int*   morp  = (const int*)d_in[0];
  const int*   seg   = (const int*)d_in[1];
  const int*   pred  = (const int*)d_in[2];
  const int*   dpin  = (const int*)d_in[3];
  const float* feat  = (const float*)d_in[4];
  const float* Wwv   = (const float*)d_in[5];
  const float* dptab = (const float*)d_in[6];
  const float* Wout  = (const float*)d_in[7];
  const float* bout  = (const float*)d_in[8];
  const float* wih0f = (const float*)d_in[9];
  const float* whh0f = (const float*)d_in[10];
  const float* b0f   = (const float*)d_in[11];
  const float* wih0b = (const float*)d_in[12];
  const float* whh0b = (const float*)d_in[13];
  const float* b0b   = (const float*)d_in[14];
  const float* wih1f = (const float*)d_in[15];
  const float* whh1f = (const float*)d_in[16];
  const float* b1f   = (const float*)d_in[17];
  const float* wih1b = (const float*)d_in[18];
  const float* whh1b = (const float*)d_in[19];
  const float* b1b   = (const float*)d_in[20];

  float* out = (float*)d_out;
  float* ws  = (float*)d_ws;

  float* tok  = ws;                        // T*EMB      = 204800
  float* inp  = tok  + (long)T_LEN * EMB;  // T*IN0      = 614400
  float* xp0f = inp  + (long)T_LEN * IN0;  // T*G4
  float* xp0b = xp0f + (long)T_LEN * G4;
  float* h1   = xp0b + (long)T_LEN * G4;   // T*IN1
  float* xp1f = h1   + (long)T_LEN * IN1;
  float* xp1b = xp1f + (long)T_LEN * G4;
  float* h2   = xp1b + (long)T_LEN * G4;   // T*IN1

  segsum_kernel<<<T_LEN, 128, 0, stream>>>(Wwv, morp, seg, tok);
  build_input_kernel<<<T_LEN, 320, 0, stream>>>(tok, pred, dptab, dpin, feat, inp);

  dim3 b32(32);
  dim3 gFull(T_LEN / 16, G4 / 16);
  gemm_wmma_f32<<<gFull, b32, 0, stream>>>(inp, wih0f, b0f, xp0f, IN0, G4, G4);
  gemm_wmma_f32<<<gFull, b32, 0, stream>>>(inp, wih0b, b0b, xp0b, IN0, G4, G4);

  lstm_scan2<<<2, 512, 0, stream>>>(xp0f, whh0f, xp0b, whh0b, h1, T_LEN);

  gemm_wmma_f32<<<gFull, b32, 0, stream>>>(h1, wih1f, b1f, xp1f, IN1, G4, G4);
  gemm_wmma_f32<<<gFull, b32, 0, stream>>>(h1, wih1b, b1b, xp1b, IN1, G4, G4);

  lstm_scan2<<<2, 512, 0, stream>>>(xp1f, whh1f, xp1b, whh1b, h2, T_LEN);

  dim3 gOut(T_LEN / 16, 64 / 16);          // N padded 50 -> 64, guarded
  gemm_wmma_f32<<<gOut, b32, 0, stream>>>(h2, Wout, bout, out, IN1, TAGS, TAGS);
}